// CreateGraphCondensation_39024072851702
// MI455X (gfx1250) — compile-verified
//
#include <hip/hip_runtime.h>
#include <math.h>

#define NB   8
#define NPE  4096
#define CD   6
#define KNN  16
#define NTOT (NB * NPE)

typedef __attribute__((ext_vector_type(2))) float v2f;
typedef __attribute__((ext_vector_type(8))) float v8f;

// ---------------------------------------------------------------------------
// Kernel 1: per-point squared norm -> ws
// ---------------------------------------------------------------------------
__global__ void sq_kernel(const float* __restrict__ coords,
                          float* __restrict__ sq, int n) {
  int i = blockIdx.x * blockDim.x + threadIdx.x;
  if (i < n) {
    const float* c = coords + (long)i * CD;
    float s = 0.f;
#pragma unroll
    for (int k = 0; k < CD; ++k) s += c[k] * c[k];
    sq[i] = s;
  }
}

// ---------------------------------------------------------------------------
// Kernel 2: per-event max score -> ws
// ---------------------------------------------------------------------------
__global__ void event_max_kernel(const float* __restrict__ score,
                                 float* __restrict__ mrss) {
  __shared__ float red[256];
  int b = blockIdx.x;
  float m = -1e30f;
  for (int i = threadIdx.x; i < NPE; i += 256)
    m = fmaxf(m, score[b * NPE + i]);
  red[threadIdx.x] = m;
  __syncthreads();
  for (int off = 128; off > 0; off >>= 1) {
    if ((int)threadIdx.x < off)
      red[threadIdx.x] = fmaxf(red[threadIdx.x], red[threadIdx.x + off]);
    __syncthreads();
  }
  if (threadIdx.x == 0) mrss[b] = red[0];
}

// ---------------------------------------------------------------------------
// Kernel 3: threshold + ordered compaction (sel_idx_up), counts -> rs_up, rs
// Single block of 1024 threads; 32 sequential chunks with LDS inclusive scan.
// ---------------------------------------------------------------------------
__global__ __launch_bounds__(1024)
void compact_kernel(const float* __restrict__ score,
                    const int* __restrict__ rs_in,
                    const float* __restrict__ mrss,
                    int* __restrict__ out_rs,
                    int* __restrict__ out_rs_up,
                    int* __restrict__ out_sel) {
  __shared__ int   lds[1024];
  __shared__ float thr_s;
  __shared__ int   base_s;
  __shared__ int   counts[NB];
  const int t = threadIdx.x;

  if (t == 0) {
    float mn = 1e30f;
    for (int e = 0; e < NB; ++e) mn = fminf(mn, mrss[e]);
    thr_s  = fminf(mn * 0.98f, 0.5f);
    base_s = 0;
    for (int e = 0; e < NB; ++e) counts[e] = 0;
  }
  __syncthreads();
  const float thr = thr_s;

  for (int chunk = 0; chunk < NTOT / 1024; ++chunk) {
    int i = chunk * 1024 + t;
    int m = (score[i] >= thr) ? 1 : 0;
    lds[t] = m;
    __syncthreads();
    for (int off = 1; off < 1024; off <<= 1) {
      int v = (t >= off) ? lds[t - off] : 0;
      __syncthreads();
      lds[t] += v;
      __syncthreads();
    }
    int incl  = lds[t];
    int total = lds[1023];
    int base  = base_s;
    if (m) out_sel[base + incl - 1] = i;
    __syncthreads();
    if (t == 0) {
      base_s = base + total;
      counts[(chunk * 1024) / NPE] += total;
    }
    __syncthreads();
  }

  const int fill = base_s;
  for (int j = fill + t; j < NTOT; j += 1024) out_sel[j] = -1;
  if (t == 0) {
    out_rs_up[0] = 0;
    int run = 0;
    for (int e = 0; e < NB; ++e) { run += counts[e]; out_rs_up[e + 1] = run; }
    for (int e = 0; e <= NB; ++e) out_rs[e] = rs_in[e];
  }
}

// ---------------------------------------------------------------------------
// Kernel 4: fused WMMA distance + top-K + softmax.
// 1 wave (32 threads) per block; block owns 32 query rows (two 16-row tiles).
// ---------------------------------------------------------------------------
__global__ __launch_bounds__(32)
void knn_kernel(const float* __restrict__ coords,
                const float* __restrict__ score,
                const float* __restrict__ sq,
                int*   __restrict__ nidx,
                float* __restrict__ dist,
                float* __restrict__ wts) {
  const int b     = blockIdx.y;
  const int qbase = blockIdx.x * 32;           // row base within event
  const int lane  = threadIdx.x;

  const float* cev = coords + (long)b * NPE * CD;
  const float* sev = score  + (long)b * NPE;
  const float* qev = sq     + (long)b * NPE;

  __shared__ float tile[32 * 17];              // padded: conflict-free banks

  // ---- A fragments (constant across candidate loop), ISA 16x4 f32 layout:
  // lanes 0-15 hold K={0,1}; lanes 16-31 hold K={2,3}; second WMMA covers K=4..7.
  const int m0 = lane & 15;
  const int kb = (lane >> 4) * 2;
  const float* q0 = cev + (long)(qbase + m0) * CD;
  const float* q1 = cev + (long)(qbase + 16 + m0) * CD;
  v2f a00 = {q0[kb], q0[kb + 1]};
  v2f a10 = {q1[kb], q1[kb + 1]};
  v2f a01 = {0.f, 0.f};
  v2f a11 = {0.f, 0.f};
  if (lane < 16) { a01.x = q0[4]; a01.y = q0[5]; a11.x = q1[4]; a11.y = q1[5]; }

  // ---- query ||x||^2 for the 8 rows this lane writes in each C/D tile
  const int rbase = (lane >> 4) * 8;
  float sqr0[8], sqr1[8];
#pragma unroll
  for (int v = 0; v < 8; ++v) {
    sqr0[v] = qev[qbase + rbase + v];
    sqr1[v] = qev[qbase + 16 + rbase + v];
  }

  // ---- private top-K (lane owns query row qbase+lane)
  float bd[KNN];
  int   bi[KNN];
#pragma unroll
  for (int k = 0; k < KNN; ++k) { bd[k] = 1e30f; bi[k] = -1; }

  const int col = lane & 15;

  for (int cb = 0; cb < NPE; cb += 16) {
    const int cand = cb + col;
    const float* cc = cev + (long)cand * CD;
    v2f bf0 = {cc[kb], cc[kb + 1]};
    v2f bf1 = {0.f, 0.f};
    if (lane < 16) { bf1.x = cc[4]; bf1.y = cc[5]; }
    const float sqc    = qev[cand];
    const bool  candok = sev[cand] > 0.5f;       // direction == 0 => candidate
    if (cb + 16 < NPE)
      __builtin_prefetch(cev + (long)(cb + 16 + col) * CD, 0, 1);

    // ---- Gram tiles: dot(query, cand) over K=8 (6 real dims + 2 zero pad)
    v8f acc0 = {};
    acc0 = __builtin_amdgcn_wmma_f32_16x16x4_f32(false, a00, false, bf0,
                                                 (short)0, acc0, false, false);
    acc0 = __builtin_amdgcn_wmma_f32_16x16x4_f32(false, a01, false, bf1,
                                                 (short)0, acc0, false, false);
    v8f acc1 = {};
    acc1 = __builtin_amdgcn_wmma_f32_16x16x4_f32(false, a10, false, bf0,
                                                 (short)0, acc1, false, false);
    acc1 = __builtin_amdgcn_wmma_f32_16x16x4_f32(false, a11, false, bf1,
                                                 (short)0, acc1, false, false);

    // ---- d2 = |q|^2 + |c|^2 - 2 dot, clamp, candidate mask; write to LDS
#pragma unroll
    for (int v = 0; v < 8; ++v) {
      float d0 = fmaxf(sqr0[v] + sqc - 2.f * acc0[v], 0.f);
      float d1 = fmaxf(sqr1[v] + sqc - 2.f * acc1[v], 0.f);
      const int r = rbase + v;
      tile[r * 17 + col]        = candok ? d0 : 1e30f;
      tile[(16 + r) * 17 + col] = candok ? d1 : 1e30f;
    }
    __syncthreads();

    // ---- branch-guarded sorted insertion into private top-16
#pragma unroll
    for (int c = 0; c < 16; ++c) {
      float v = tile[lane * 17 + c];
      if (v < bd[KNN - 1]) {
        int ci = cb + c;
#pragma unroll
        for (int k = 0; k < KNN; ++k) {
          const bool  lt = v < bd[k];
          const float td = bd[k];
          const int   ti = bi[k];
          bd[k] = lt ? v  : td;
          bi[k] = lt ? ci : ti;
          v     = lt ? td : v;
          ci    = lt ? ti : ci;
        }
      }
    }
    __syncthreads();
  }

  // ---- epilogue: validity mask, global indices, softmax(exp(-dist))
  const int  grow = b * NPE + qbase + lane;
  const bool qv   = sev[qbase + lane] <= 0.5f;   // direction == 1
  float tv[KNN];
  float mx = -1e30f;
#pragma unroll
  for (int k = 0; k < KNN; ++k) {
    const bool valid = qv && (bd[k] < 5e29f);
    const int  ni    = valid ? (bi[k] + b * NPE) : -1;
    const float dk   = valid ? bd[k] : 0.f;
    nidx[(long)grow * KNN + k] = ni;
    dist[(long)grow * KNN + k] = dk;
    tv[k] = __expf(-dk);
    mx = fmaxf(mx, tv[k]);
  }
  float s = 0.f;
#pragma unroll
  for (int k = 0; k < KNN; ++k) { tv[k] = __expf(tv[k] - mx); s += tv[k]; }
  const float inv = 1.f / s;
#pragma unroll
  for (int k = 0; k < KNN; ++k) wts[(long)grow * KNN + k] = tv[k] * inv;
}

// ---------------------------------------------------------------------------
extern "C" void kernel_launch(void* const* d_in, const int* in_sizes, int n_in,
                              void* d_out, int out_size, void* d_ws,
                              size_t ws_size, hipStream_t stream) {
  const float* score  = (const float*)d_in[0];   // (32768,1) f32
  const float* coords = (const float*)d_in[1];   // (32768,6) f32
  const int*   rs     = (const int*)d_in[2];     // (9,) i32

  // Output tuple concatenated flat: rs(9) rs_up(9) sel(32768)
  //                                 nidx(32768*16) dist(32768*16) w(32768*16)
  float* out       = (float*)d_out;
  int*   out_rs    = (int*)out;
  int*   out_rs_up = out_rs + 9;
  int*   out_sel   = out_rs + 18;
  int*   out_nidx  = out_rs + 18 + NTOT;
  float* out_dist  = out + 18 + NTOT + (long)NTOT * KNN;
  float* out_w     = out_dist + (long)NTOT * KNN;

  float* ws_sq   = (float*)d_ws;       // NTOT floats
  float* ws_mrss = ws_sq + NTOT;       // NB floats

  sq_kernel<<<(NTOT + 255) / 256, 256, 0, stream>>>(coords, ws_sq, NTOT);
  event_max_kernel<<<NB, 256, 0, stream>>>(score, ws_mrss);
  compact_kernel<<<1, 1024, 0, stream>>>(score, rs, ws_mrss,
                                         out_rs, out_rs_up, out_sel);
  knn_kernel<<<dim3(NPE / 32, NB), 32, 0, stream>>>(coords, score, ws_sq,
                                                    out_nidx, out_dist, out_w);
}